// GCN_45518063403696
// MI455X (gfx1250) — compile-verified
//
#include <hip/hip_runtime.h>
#include <math.h>

// ---------------------------------------------------------------------------
// GAT stack for gfx1250 (MI455X). Dense linears: V_WMMA_F32_16X16X4_F32 with
// fully-unrolled compile-time K loop, branchless b64 fragment loads (zero-
// padded operands, transposed-padded weights). Edge softmax/aggregation uses
// native u32/f32 global atomics. All launches on `stream`, no alloc/sync.
// Assumes N % 16 == 0 (N = 50000).
// ---------------------------------------------------------------------------

typedef __attribute__((ext_vector_type(2))) float v2f;
typedef __attribute__((ext_vector_type(8))) float v8f;

#define NEG_SLOPE 0.2f

// monotone float <-> uint mapping for atomicMax-based segment max
__device__ __forceinline__ unsigned fenc(float f) {
  unsigned u = __float_as_uint(f);
  return (u & 0x80000000u) ? ~u : (u | 0x80000000u);
}
__device__ __forceinline__ float fdec(unsigned u) {
  return (u & 0x80000000u) ? __uint_as_float(u & 0x7FFFFFFFu)
                           : __uint_as_float(~u);
}

// ---------------------------------------------------------------------------
// Pad + transpose weights: WPT[c][k] = W[k][c] for k<din,c<dout, else 0.
// WPT is [doutp x dinp] row-major so the B fragment (two consecutive K values
// of one output column) is a single aligned b64 load.
// ---------------------------------------------------------------------------
__global__ void pad_wT(const float* __restrict__ W, float* __restrict__ WPT,
                       int din, int dout, int dinp, int doutp) {
  int i = blockIdx.x * blockDim.x + threadIdx.x;
  if (i >= dinp * doutp) return;
  int c = i / dinp;
  int k = i - c * dinp;
  WPT[i] = (k < din && c < dout) ? W[k * dout + c] : 0.0f;
}

// ---------------------------------------------------------------------------
// Y[n, 0:doutp) = X[n, 0:dinp) @ WPT^T ; one wave per 16x16 tile.
// A frag (16x4 f32): lane l<16, half h: {A[m=l][2h], A[m=l][2h+1]}
// B frag (4x16 f32): lane l<16, half h: {B[2h][n=l], B[2h+1][n=l]}
// D (16x16 f32):     vgpr r, half h:    D[r+8h][n=lane%16]
// Requires: nrows % 16 == 0, X padded cols [din,dinp) zero, ld_in >= dinp.
// ---------------------------------------------------------------------------
template <int DINP>
__global__ void linear_wmma_f32(const float* __restrict__ X,
                                const float* __restrict__ WPT,
                                float* __restrict__ Y,
                                int tilesM, int ld_in, int doutp, int tilesN) {
  int wave = (blockIdx.x * blockDim.x + threadIdx.x) >> 5;
  int lane = threadIdx.x & 31;
  if (wave >= tilesM * tilesN) return;   // wave-uniform: EXEC stays all-ones

  int tm = wave / tilesN;
  int tn = wave - tm * tilesN;
  int row0 = tm << 4;
  int col0 = tn << 4;
  int half = lane >> 4;                  // 0 or 1
  int l    = lane & 15;
  int arow = row0 + l;                   // A-matrix M index
  int bcol = col0 + l;                   // B-matrix N index

  const float* Ap = X   + (size_t)arow * ld_in + 2 * half;
  const float* Bp = WPT + (size_t)bcol * DINP  + 2 * half;

  v8f acc = {};
  #pragma unroll
  for (int k0 = 0; k0 < DINP; k0 += 4) {
    v2f a = *reinterpret_cast<const v2f*>(Ap + k0);
    v2f b = *reinterpret_cast<const v2f*>(Bp + k0);
    acc = __builtin_amdgcn_wmma_f32_16x16x4_f32(
        false, a, false, b, (short)0, acc, false, false);
  }

  float* Yp = Y + (size_t)(row0 + 8 * half) * doutp + col0 + l;
  #pragma unroll
  for (int r = 0; r < 8; ++r)
    Yp[(size_t)r * doutp] = acc[r];
}

// per-node attention coefficients: as[n] = xs[n,:]·a_src, ad[n] = xs[n,:]·a_dst
__global__ void attn_coef(const float* __restrict__ XS,
                          const float* __restrict__ a_src,
                          const float* __restrict__ a_dst,
                          float* __restrict__ as_, float* __restrict__ ad_,
                          int nrows, int dout, int ldxs) {
  int n = blockIdx.x * blockDim.x + threadIdx.x;
  if (n >= nrows) return;
  const float* row = XS + (size_t)n * ldxs;
  float s = 0.0f, d = 0.0f;
  for (int c = 0; c < dout; ++c) {
    float v = row[c];
    s += v * a_src[c];
    d += v * a_dst[c];
  }
  as_[n] = s;
  ad_[n] = d;
}

__global__ void init_agg(float* __restrict__ AGG, unsigned* __restrict__ mxu,
                         float* __restrict__ den, int nrows, int dout) {
  int i = blockIdx.x * blockDim.x + threadIdx.x;
  if (i < nrows * dout) AGG[i] = 0.0f;
  if (i < nrows) { mxu[i] = 0u; den[i] = 0.0f; }
}

// pass 1: logits + segment max over dst
__global__ void edge_logits(const int* __restrict__ src,
                            const int* __restrict__ dst, int E_, int nrows,
                            const float* __restrict__ as_,
                            const float* __restrict__ ad_,
                            float* __restrict__ lg, unsigned* __restrict__ mxu) {
  int e = blockIdx.x * blockDim.x + threadIdx.x;
  if (e >= E_ + nrows) return;
  int s = (e < E_) ? src[e] : (e - E_);
  int d = (e < E_) ? dst[e] : (e - E_);
  float z = as_[s] + ad_[d];
  z = (z > 0.0f) ? z : NEG_SLOPE * z;        // leaky_relu
  lg[e] = z;
  atomicMax(&mxu[d], fenc(z));
}

// pass 2: exp(logit - max) + segment sum
__global__ void edge_expsum(const int* __restrict__ src,
                            const int* __restrict__ dst, int E_, int nrows,
                            const float* __restrict__ lg,
                            const unsigned* __restrict__ mxu,
                            float* __restrict__ exb, float* __restrict__ den) {
  int e = blockIdx.x * blockDim.x + threadIdx.x;
  if (e >= E_ + nrows) return;
  int d = (e < E_) ? dst[e] : (e - E_);
  float ex = expf(lg[e] - fdec(mxu[d]));
  exb[e] = ex;
  atomicAdd(&den[d], ex);
}

// pass 3: AGG[d,:] += alpha * xs[s,:]; 4 threads cooperate per edge
__global__ void edge_aggregate(const int* __restrict__ src,
                               const int* __restrict__ dst, int E_, int nrows,
                               const float* __restrict__ exb,
                               const float* __restrict__ den,
                               const float* __restrict__ XS,
                               float* __restrict__ AGG, int dout, int ldxs) {
  int t = blockIdx.x * blockDim.x + threadIdx.x;
  int e = t >> 2;
  int sub = t & 3;
  if (e >= E_ + nrows) return;
  int s = (e < E_) ? src[e] : (e - E_);
  int d = (e < E_) ? dst[e] : (e - E_);
  float alpha = exb[e] / den[d];
  const float* xr = XS + (size_t)s * ldxs;
  float* orow = AGG + (size_t)d * dout;
  for (int c = sub; c < dout; c += 4)
    atomicAdd(&orow[c], alpha * xr[c]);
}

// out[n, c] = relu?(AGG[n,c] + b[c]) for c<dout, 0 for c in [dout, ld_out)
__global__ void finalize_k(const float* __restrict__ AGG,
                           const float* __restrict__ bias,
                           float* __restrict__ out, int nrows, int dout,
                           int ld_out, int do_relu) {
  int i = blockIdx.x * blockDim.x + threadIdx.x;
  if (i >= nrows * ld_out) return;
  int c = i % ld_out;
  float v = 0.0f;
  if (c < dout) {
    int n = i / ld_out;
    v = AGG[(size_t)n * dout + c] + bias[c];
    if (do_relu) v = fmaxf(v, 0.0f);
  }
  out[i] = v;
}

// out[n] = {a[n], b[n], 0, 0}  (stride 4, zero-padded for K padding)
__global__ void concat2p(const float* __restrict__ a, const float* __restrict__ b,
                         float* __restrict__ out, int nrows) {
  int n = blockIdx.x * blockDim.x + threadIdx.x;
  if (n >= nrows) return;
  out[4 * n + 0] = a[n];
  out[4 * n + 1] = b[n];
  out[4 * n + 2] = 0.0f;
  out[4 * n + 3] = 0.0f;
}

__global__ void relu_copy(const float* __restrict__ in, float* __restrict__ out,
                          int n) {
  int i = blockIdx.x * blockDim.x + threadIdx.x;
  if (i >= n) return;
  out[i] = fmaxf(in[i], 0.0f);
}

// ---------------------------------------------------------------------------
extern "C" void kernel_launch(void* const* d_in, const int* in_sizes, int n_in,
                              void* d_out, int out_size, void* d_ws,
                              size_t ws_size, hipStream_t stream) {
  const float* x  = (const float*)d_in[0];
  const int*   ei = (const int*)d_in[1];
  const float* m  = (const float*)d_in[2];
  const float* f  = (const float*)d_in[3];

  const int N    = in_sizes[2];         // m has N elements
  const int E    = in_sizes[1] / 2;     // edge_index is [2, E]
  const int FIN  = in_sizes[0] / N;     // 8
  const int Etot = E + N;
  const int* src = ei;
  const int* dst = ei + E;

  // params per layer (in _LAYER_DIMS insertion order): W, a_src, a_dst, b
  struct P { const float *W, *as, *ad, *b; };
  P prm[10];
  for (int i = 0; i < 10; ++i) {
    prm[i].W  = (const float*)d_in[4 + 4 * i + 0];
    prm[i].as = (const float*)d_in[4 + 4 * i + 1];
    prm[i].ad = (const float*)d_in[4 + 4 * i + 2];
    prm[i].b  = (const float*)d_in[4 + 4 * i + 3];
  }

  // workspace carve-out (floats)
  float* ws = (float*)d_ws;
  size_t o = 0;
  float*    XS  = ws + o; o += (size_t)N * 112;  // linear output, padded stride
  float*    AGG = ws + o; o += (size_t)N * 100;  // compact aggregation buffer
  float*    H   = ws + o; o += (size_t)N * 100;  // [N,100] hidden
  float*    AS  = ws + o; o += N;
  float*    AD  = ws + o; o += N;
  unsigned* MXU = (unsigned*)(ws + o); o += N;
  float*    DEN = ws + o; o += N;
  float*    LG  = ws + o; o += Etot;
  float*    EXB = ws + o; o += Etot;
  float*    B2  = ws + o; o += N;                // [N,1]
  float*    B10 = ws + o; o += (size_t)N * 12;   // [N,10] padded to 12
  float*    HM1 = ws + o; o += N;
  float*    HF1 = ws + o; o += N;
  float*    CIN = ws + o; o += (size_t)N * 4;    // [N,2] padded to 4
  float*    WPT = ws + o; o += (size_t)112 * 112; // padded transposed weights

  const int TB = 256;
  auto blocks = [](long n, int tb) { return (unsigned)((n + tb - 1) / tb); };

  // one GAT conv; ld_in >= dinp with zeroed pad columns, ld_out gets zero pads
  auto conv = [&](const float* hin, int ld_in, int din, int dout, const P& p,
                  bool relu, float* hout, int ld_out) {
    int dinp  = (din + 3) & ~3;
    int doutp = (dout + 15) & ~15;
    int tilesM = N / 16;
    int tilesN = doutp / 16;
    long waves = (long)tilesM * tilesN;

    pad_wT<<<blocks(dinp * doutp, TB), TB, 0, stream>>>(p.W, WPT, din, dout,
                                                        dinp, doutp);
    switch (dinp) {
      case 4:
        linear_wmma_f32<4><<<blocks(waves * 32, TB), TB, 0, stream>>>(
            hin, WPT, XS, tilesM, ld_in, doutp, tilesN);
        break;
      case 8:
        linear_wmma_f32<8><<<blocks(waves * 32, TB), TB, 0, stream>>>(
            hin, WPT, XS, tilesM, ld_in, doutp, tilesN);
        break;
      case 12:
        linear_wmma_f32<12><<<blocks(waves * 32, TB), TB, 0, stream>>>(
            hin, WPT, XS, tilesM, ld_in, doutp, tilesN);
        break;
      default:
        linear_wmma_f32<100><<<blocks(waves * 32, TB), TB, 0, stream>>>(
            hin, WPT, XS, tilesM, ld_in, doutp, tilesN);
        break;
    }
    attn_coef<<<blocks(N, TB), TB, 0, stream>>>(XS, p.as, p.ad, AS, AD, N, dout,
                                                doutp);
    init_agg<<<blocks((long)N * dout, TB), TB, 0, stream>>>(AGG, MXU, DEN, N,
                                                            dout);
    edge_logits<<<blocks(Etot, TB), TB, 0, stream>>>(src, dst, E, N, AS, AD, LG,
                                                     MXU);
    edge_expsum<<<blocks(Etot, TB), TB, 0, stream>>>(src, dst, E, N, LG, MXU,
                                                     EXB, DEN);
    edge_aggregate<<<blocks((long)Etot * 4, TB), TB, 0, stream>>>(
        src, dst, E, N, EXB, DEN, XS, AGG, dout, doutp);
    finalize_k<<<blocks((long)N * ld_out, TB), TB, 0, stream>>>(
        AGG, p.b, hout, N, dout, ld_out, relu ? 1 : 0);
  };

  float* OUT_h  = (float*)d_out;   // h      -> [0, N)
  float* OUT_hm = OUT_h + N;       // hm_br  -> [N, 2N)
  float* OUT_hf = OUT_h + 2 * N;   // hf_br  -> [2N, 3N)

  // --- path 1 (m) ---
  conv(x,   FIN, FIN, 100, prm[0], true,  H,      100);  // conv1 + relu
  conv(H,   100, 100, 1,   prm[2], false, B2,     1);    // conv2
  concat2p<<<blocks(N, TB), TB, 0, stream>>>(B2, m, CIN, N);
  conv(CIN, 4,   2,   10,  prm[4], true,  B10,    12);   // conv2m + relu
  conv(B10, 12,  10,  1,   prm[5], false, OUT_hm, 1);    // conv2m_1 -> hm_br
  relu_copy<<<blocks(N, TB), TB, 0, stream>>>(OUT_hm, HM1, N);

  // --- path 2 (f) ---
  conv(x,   FIN, FIN, 100, prm[1], true,  H,      100);  // conv1_2 + relu
  conv(H,   100, 100, 1,   prm[3], false, B2,     1);    // conv2_2
  concat2p<<<blocks(N, TB), TB, 0, stream>>>(B2, f, CIN, N);
  conv(CIN, 4,   2,   10,  prm[6], true,  B10,    12);   // conv2f + relu
  conv(B10, 12,  10,  1,   prm[7], false, OUT_hf, 1);    // conv2f_1 -> hf_br
  relu_copy<<<blocks(N, TB), TB, 0, stream>>>(OUT_hf, HF1, N);

  // --- merge ---
  concat2p<<<blocks(N, TB), TB, 0, stream>>>(HM1, HF1, CIN, N);
  conv(CIN, 4,   2,   10,  prm[8], true,  B10,    12);   // fc_A + relu
  conv(B10, 12,  10,  1,   prm[9], false, OUT_h,  1);    // fcA_1 -> h
}